// LabelSmoothing_755914244560
// MI455X (gfx1250) — compile-verified
//
#include <hip/hip_runtime.h>

typedef float v2f __attribute__((ext_vector_type(2)));
typedef float v4f __attribute__((ext_vector_type(4)));
typedef float v8f __attribute__((ext_vector_type(8)));

#define BLK 256
#define SMOOTHING 0.1f

// One block per row of x[N, V]. Streams the row with unconditional b128
// non-temporal loads (guard-free main loop -> load clauses, staged waits),
// feeds values through V_WMMA_F32_16X16X4_F32 with an all-ones B matrix so
// the matrix pipe performs the accumulation: D[m][n] += sum_k A[m][k], and
// the grand total of all 256 D elements == 16 * sum of all A inputs.
// Four independent accumulators keep WMMAs off each other's D->C chains.
__global__ void __launch_bounds__(BLK)
ls_row_reduce(const float* __restrict__ x,
              const long long* __restrict__ target,
              float* __restrict__ part_sum,
              float* __restrict__ part_gather,
              int vdim) {
    const int row = blockIdx.x;
    const int tid = threadIdx.x;
    const int v4 = vdim >> 2;                 // float4s per row (8000)
    const int nfull = v4 / BLK;               // guard-free iterations (31)
    const float* rowp = x + (size_t)row * (size_t)vdim;
    const v4f* row4 = (const v4f*)rowp;

    v2f ones;
    ones.x = 1.0f;
    ones.y = 1.0f;

    v8f acc0 = {0.f, 0.f, 0.f, 0.f, 0.f, 0.f, 0.f, 0.f};
    v8f acc1 = acc0, acc2 = acc0, acc3 = acc0;

    // Main loop: all lanes in-bounds, no EXEC manipulation, loads can clause.
    int it = 0;
    for (; it + 1 < nfull; it += 2) {
        v4f va = __builtin_nontemporal_load(&row4[tid + (it + 0) * BLK]);
        v4f vb = __builtin_nontemporal_load(&row4[tid + (it + 1) * BLK]);
        v2f a0, a1, b0, b1;
        a0.x = va.x; a0.y = va.y;
        a1.x = va.z; a1.y = va.w;
        b0.x = vb.x; b0.y = vb.y;
        b1.x = vb.z; b1.y = vb.w;
        acc0 = __builtin_amdgcn_wmma_f32_16x16x4_f32(false, a0, false, ones, (short)0, acc0, false, false);
        acc1 = __builtin_amdgcn_wmma_f32_16x16x4_f32(false, a1, false, ones, (short)0, acc1, false, false);
        acc2 = __builtin_amdgcn_wmma_f32_16x16x4_f32(false, b0, false, ones, (short)0, acc2, false, false);
        acc3 = __builtin_amdgcn_wmma_f32_16x16x4_f32(false, b1, false, ones, (short)0, acc3, false, false);
    }
    for (; it < nfull; ++it) {
        v4f v = __builtin_nontemporal_load(&row4[tid + it * BLK]);
        v2f a0, a1;
        a0.x = v.x; a0.y = v.y;
        a1.x = v.z; a1.y = v.w;
        acc0 = __builtin_amdgcn_wmma_f32_16x16x4_f32(false, a0, false, ones, (short)0, acc0, false, false);
        acc1 = __builtin_amdgcn_wmma_f32_16x16x4_f32(false, a1, false, ones, (short)0, acc1, false, false);
    }

    // Tail: zero-fill out-of-range lanes so WMMA still runs with EXEC==~0.
    {
        const int j = nfull * BLK + tid;
        v4f v = {0.f, 0.f, 0.f, 0.f};
        if (j < v4) v = __builtin_nontemporal_load(&row4[j]);
        v2f a0, a1;
        a0.x = v.x; a0.y = v.y;
        a1.x = v.z; a1.y = v.w;
        acc2 = __builtin_amdgcn_wmma_f32_16x16x4_f32(false, a0, false, ones, (short)0, acc2, false, false);
        acc3 = __builtin_amdgcn_wmma_f32_16x16x4_f32(false, a1, false, ones, (short)0, acc3, false, false);
    }

    // Per-lane sum of the 8 accumulator VGPRs; across the wave's 32 lanes this
    // covers all 256 D elements (== 16x the true input sum).
    v8f a = (acc0 + acc1) + (acc2 + acc3);
    float s = ((a[0] + a[1]) + (a[2] + a[3])) + ((a[4] + a[5]) + (a[6] + a[7]));

    __shared__ float red[BLK];
    red[tid] = s;
    __syncthreads();
    #pragma unroll
    for (int off = BLK / 2; off > 0; off >>= 1) {
        if (tid < off) red[tid] += red[tid + off];
        __syncthreads();
    }

    if (tid == 0) {
        const long long t = target[row];
        const float maskf = (t != 0) ? 1.0f : 0.0f;   // PADDING_IDX == 0
        part_sum[row]    = (red[0] * (1.0f / 16.0f)) * maskf;
        part_gather[row] = rowp[t] * maskf;
    }
}

// Single block: deterministic fixed-order reduction of the per-row partials.
__global__ void __launch_bounds__(BLK)
ls_finalize(const float* __restrict__ part_sum,
            const float* __restrict__ part_gather,
            float* __restrict__ out,
            int nrows, int vdim) {
    const int tid = threadIdx.x;
    float ss = 0.f, gg = 0.f;
    for (int i = tid; i < nrows; i += BLK) {
        ss += part_sum[i];
        gg += part_gather[i];
    }
    __shared__ float rs[BLK];
    __shared__ float rg[BLK];
    rs[tid] = ss;
    rg[tid] = gg;
    __syncthreads();
    #pragma unroll
    for (int off = BLK / 2; off > 0; off >>= 1) {
        if (tid < off) {
            rs[tid] += rs[tid + off];
            rg[tid] += rg[tid + off];
        }
        __syncthreads();
    }
    if (tid == 0) {
        const float loss = -rs[0];
        const float nll  = -rg[0];
        out[0] = SMOOTHING * (loss / (float)vdim) + (1.0f - SMOOTHING) * nll;
    }
}

extern "C" void kernel_launch(void* const* d_in, const int* in_sizes, int n_in,
                              void* d_out, int out_size, void* d_ws, size_t ws_size,
                              hipStream_t stream) {
    (void)n_in; (void)out_size; (void)ws_size;

    const float* x          = (const float*)d_in[0];
    const long long* target = (const long long*)d_in[1];
    float* out              = (float*)d_out;

    const int nrows = in_sizes[1];            // 4096
    const int vdim  = in_sizes[0] / nrows;    // 32000

    float* part_sum    = (float*)d_ws;        // [nrows]
    float* part_gather = part_sum + nrows;    // [nrows]

    ls_row_reduce<<<nrows, BLK, 0, stream>>>(x, target, part_sum, part_gather, vdim);
    ls_finalize<<<1, BLK, 0, stream>>>(part_sum, part_gather, out, nrows, vdim);
}